// ExponentialUnitNorm_50242527428868
// MI455X (gfx1250) — compile-verified
//
#include <hip/hip_runtime.h>

// ExponentialUnitNorm for MI455X (gfx1250).
// EMA scan over T expressed as blocked 16x16x16 matmuls on the WMMA pipe:
//   S_block = W * M_block + p (outer) s_carry,  W[j][k] = (1-a)*a^(j-k) (lower tri)
// One wave per (b, c, f-tile16) chain: 16*2*31 = 992 waves, 125 blocks of 16 t each.
// Raw v_sqrt/v_rsq (inputs clamped >= 1e-14, so ~1ulp approx is fine),
// 4-float2 SWAPX16 exchange, distance-2 load pipeline.

#define ALPHA 0.99f
#define EPSQ  1e-14f

typedef __attribute__((ext_vector_type(2))) float v2f;
typedef __attribute__((ext_vector_type(8))) float v8f;

__device__ __forceinline__ float powa(int n) {
  float v = 1.0f;
  #pragma unroll 1
  for (int i = 0; i < n; ++i) v *= ALPHA;
  return v;
}

__global__ __launch_bounds__(256) void eun_wmma_kernel(
    const float* __restrict__ x, const float* __restrict__ s0,
    float* __restrict__ y) {
  const int lane = threadIdx.x & 31;
  const int wid  = (blockIdx.x * (int)blockDim.x + (int)threadIdx.x) >> 5; // 0..991
  const int NT = 31;                       // ceil(481/16) f-tiles
  const int ftile = wid % NT;
  const int c     = (wid / NT) & 1;
  const int b     = wid / (NT * 2);

  const int h = lane >> 4;                 // half-wave (0/1)
  const int n = lane & 15;                 // column (f within tile) and A-row M
  const int f = ftile * 16 + n;
  const bool valid = (f < 481);
  const int fc = valid ? f : 480;          // clamp for in-bounds loads

  const long rowstride = 481L * 2L;        // floats per (b,c,t) row
  const long base = ((long)(b * 2 + c)) * 2000L * rowstride;

  // Per-lane load/store base pointers (advance by 16 rows per block).
  const float2* ldp = reinterpret_cast<const float2*>(x + base + 2L * fc);
  float2*       stp = reinterpret_cast<float2*>(y + base + 2L * f);
  const long blkstep = 16L * 481L;         // float2 units per 16-t block

  // ---- A operands: W split into four 16x4 K-chunks.
  // A layout (16x4 f32): M = lane&15; VGPR0 holds K = 4q+2h, VGPR1 holds K = 4q+2h+1.
  v2f A[4];
  #pragma unroll
  for (int q = 0; q < 4; ++q) {
    const int k0 = 4 * q + 2 * h;
    A[q].x = (k0     <= n) ? (1.0f - ALPHA) * powa(n - k0)     : 0.0f;
    A[q].y = (k0 + 1 <= n) ? (1.0f - ALPHA) * powa(n - k0 - 1) : 0.0f;
  }
  // Carry-injection weights: C/D row M = v + 8h, weight = alpha^(M+1).
  float pw[8];
  #pragma unroll
  for (int v = 0; v < 8; ++v) pw[v] = powa(v + 8 * h + 1);

  float carry = s0[fc];                    // s_{-1} per column

  // B-operand row set per lane: slot i = 2q+j -> t-offset 4q + 2h + j within block.
  long roff[8];
  #pragma unroll
  for (int q = 0; q < 4; ++q)
    #pragma unroll
    for (int j = 0; j < 2; ++j)
      roff[2 * q + j] = (long)(4 * q + 2 * h + j) * 481L;   // float2 units

  // Distance-2 software pipeline: three live buffers.
  float2 b0[8], b1[8], b2[8];
  #pragma unroll
  for (int i = 0; i < 8; ++i) b0[i] = ldp[roff[i]];
  #pragma unroll
  for (int i = 0; i < 8; ++i) b1[i] = ldp[blkstep + roff[i]];

  for (int blk = 0; blk < 125; ++blk) {
    if (blk + 2 < 125) {
      const long bo = (long)(blk + 2) * blkstep;
      #pragma unroll
      for (int i = 0; i < 8; ++i) b2[i] = ldp[bo + roff[i]];
    }

    // Magnitudes m = sqrt(max(re^2 + im^2, EPS)) -- raw v_sqrt_f32 (arg >= 1e-14).
    float m[8];
    #pragma unroll
    for (int i = 0; i < 8; ++i)
      m[i] = __builtin_amdgcn_sqrtf(
          fmaxf(fmaf(b0[i].x, b0[i].x, b0[i].y * b0[i].y), EPSQ));

    // Accumulator init: C[j][f] = alpha^(j+1) * carry[f]
    v8f acc;
    #pragma unroll
    for (int v = 0; v < 8; ++v) acc[v] = pw[v] * carry;

    // Four chained WMMAs: S += W[:,4q:4q+4] * M[4q:4q+4,:]
    #pragma unroll
    for (int q = 0; q < 4; ++q) {
      v2f bq; bq.x = m[2 * q]; bq.y = m[2 * q + 1];
      acc = __builtin_amdgcn_wmma_f32_16x16x4_f32(
          false, A[q], false, bq, (short)0, acc, false, false);
    }

    // Carry = row 15 = VGPR7 of lanes 16..31; broadcast per column.
    carry = __shfl(acc[7], 16 + n, 32);

    // Minimal half-wave exchange: each half sends the 4 float2 the other needs.
    // h=0 sends slots 4..7 (rows 8,9,12,13), h=1 sends slots 0..3 (rows 2,3,6,7).
    float2 ex[4];
    #pragma unroll
    for (int i = 0; i < 4; ++i) {
      float2 snd = h ? b0[i] : b0[4 + i];
      ex[i].x = __shfl_xor(snd.x, 16, 32);
      ex[i].y = __shfl_xor(snd.y, 16, 32);
    }
    // Output rows v+8h for v=0..7:
    //  h=0: rows 0..7  = own{0,1,4,5} + ex{2,3,6,7}
    //  h=1: rows 8..15 = ex{8,9,12,13} + own{10,11,14,15}
    float2 xo[8];
    xo[0] = h ? ex[0]  : b0[0];
    xo[1] = h ? ex[1]  : b0[1];
    xo[2] = h ? b0[4]  : ex[0];
    xo[3] = h ? b0[5]  : ex[1];
    xo[4] = h ? ex[2]  : b0[2];
    xo[5] = h ? ex[3]  : b0[3];
    xo[6] = h ? b0[6]  : ex[2];
    xo[7] = h ? b0[7]  : ex[3];

    // y = x * rsqrt(s); row for v is blk*16 + v + 8h. Stores predicated on f<481.
    if (valid) {
      const long so = (long)blk * blkstep + (long)(8 * h) * 481L;
      #pragma unroll
      for (int v = 0; v < 8; ++v) {
        const float inv = __builtin_amdgcn_rsqf(acc[v]);
        float2 o; o.x = xo[v].x * inv; o.y = xo[v].y * inv;
        stp[so + (long)v * 481L] = o;
      }
    }

    #pragma unroll
    for (int i = 0; i < 8; ++i) { b0[i] = b1[i]; b1[i] = b2[i]; }
  }
}

extern "C" void kernel_launch(void* const* d_in, const int* in_sizes, int n_in,
                              void* d_out, int out_size, void* d_ws, size_t ws_size,
                              hipStream_t stream) {
  (void)in_sizes; (void)n_in; (void)d_ws; (void)ws_size; (void)out_size;
  const float* x  = (const float*)d_in[0];   // [16,2,2000,481,2] f32
  const float* s0 = (const float*)d_in[1];   // [481] f32
  float* y = (float*)d_out;

  // 992 chains, 8 waves (256 threads) per block -> 124 blocks.
  eun_wmma_kernel<<<124, 256, 0, stream>>>(x, s0, y);
}